// MessagePassing_43997644980995
// MI455X (gfx1250) — compile-verified
//
#include <hip/hip_runtime.h>

// Problem constants (match reference)
#define NN 100000
#define EE 1600000
#define DD 128
#define HSTRIDE 132   // padded LDS row stride (floats) for hidden tile: 4*m16 bank spread
#define WPAIR   288   // dwords per K-pair row of swizzled W; +WPAIR lands banks 32..63

typedef __attribute__((ext_vector_type(2))) float v2f;
typedef __attribute__((ext_vector_type(8))) float v8f;

// ---------------------------------------------------------------------------
// Fused 2-layer MLP: Y = relu(X @ W1 + b1) @ W2 + b2
// One 128-row tile per block (8 wave32s). Full fp32 via V_WMMA_F32_16X16X4_F32.
// Wave w computes rows [w*16, w*16+16) x all 128 cols (8 col-tiles of 16).
//
// WMMA f32 16x16x4 fragment layouts (ISA 7.12.2):
//   A (16x4):  lane<16 -> row M=lane, K = k0+0..1 ; lane>=16 -> row M=lane-16, K = k0+2..3
//   B (4x16):  lane<16 -> col N=lane, rows K=k0 (v0), k0+1 (v1); lane>=16 -> K=k0+2, k0+3
//   C/D 16x16: vgpr i -> row M = i + (lane>=16 ? 8:0), col N = lane%16
//
// W is staged pair-swizzled: sW[p*WPAIR + 2n + q] = W[2p+q][n], so a B fragment
// is one aligned ds_load_b64 (no cross-register assembly, no bank conflicts).
// ---------------------------------------------------------------------------
__global__ __launch_bounds__(256) void mlp_fused_kernel(
    const float* __restrict__ X,
    const float* __restrict__ W1, const float* __restrict__ b1,
    const float* __restrict__ W2, const float* __restrict__ b2,
    float* __restrict__ Y, int nRows)
{
    __shared__ float sW[(DD / 2) * WPAIR];  // 73.7 KB: swizzled W1, then W2
    __shared__ float sH[DD * HSTRIDE];      // 67.6 KB: hidden tile (padded)

    const int tid  = threadIdx.x;
    const int wave = tid >> 5;
    const int lane = tid & 31;
    const int m16  = lane & 15;
    const int half = lane >> 4;          // selects K sub-pair / row half
    const int row0 = blockIdx.x * DD;    // 128-row tile base

    // ---- stage W1 into LDS, pair-swizzled ----
    {
        const float2* src = (const float2*)W1;
        #pragma unroll 4
        for (int i = tid; i < DD * DD / 2; i += 256) {
            float2 t2 = src[i];
            const int k = (i * 2) >> 7;      // row of W
            const int n = (i * 2) & 127;     // col of W (even)
            float* dst = &sW[(k >> 1) * WPAIR + (k & 1)];
            dst[n * 2]       = t2.x;
            dst[(n + 1) * 2] = t2.y;
        }
    }
    __syncthreads();

    // Row clamp instead of predication: a clamped (duplicate) A row only feeds
    // the same out-of-range output row, and output stores are guarded below.
    const int  arow = row0 + wave * 16 + m16;
    const int  crow = arow < nRows ? arow : (nRows - 1);
    const float* xrow = X + (size_t)crow * DD;

    // ---- GEMM1: hidden = relu(X @ W1 + b1) ----
    {
        v8f acc[8] = {};
        for (int k0 = 0; k0 < DD; k0 += 4) {
            const int kk = k0 + half * 2;        // even
            float2 t2 = *(const float2*)(xrow + kk);
            v2f a; a.x = t2.x; a.y = t2.y;
            const float* bbase = &sW[(kk >> 1) * WPAIR + m16 * 2];
            #pragma unroll
            for (int t = 0; t < 8; ++t) {
                v2f b = *(const v2f*)(bbase + t * 32);
                acc[t] = __builtin_amdgcn_wmma_f32_16x16x4_f32(
                    false, a, false, b, (short)0, acc[t], false, false);
            }
        }
        // relu + b1 -> LDS hidden tile
        const int hr = wave * 16 + half * 8;
        #pragma unroll
        for (int t = 0; t < 8; ++t) {
            const int col  = t * 16 + m16;
            const float bb = b1[col];
            #pragma unroll
            for (int i = 0; i < 8; ++i) {
                float h = acc[t][i] + bb;
                sH[(hr + i) * HSTRIDE + col] = h > 0.f ? h : 0.f;
            }
        }
    }
    __syncthreads();   // all waves done reading W1 & writing hidden

    // ---- stage W2 into LDS (reuse sW), pair-swizzled ----
    {
        const float2* src = (const float2*)W2;
        #pragma unroll 4
        for (int i = tid; i < DD * DD / 2; i += 256) {
            float2 t2 = src[i];
            const int k = (i * 2) >> 7;
            const int n = (i * 2) & 127;
            float* dst = &sW[(k >> 1) * WPAIR + (k & 1)];
            dst[n * 2]       = t2.x;
            dst[(n + 1) * 2] = t2.y;
        }
    }
    __syncthreads();

    // ---- GEMM2: Y = hidden @ W2 + b2 ----
    {
        v8f acc[8] = {};
        const float* hrow = &sH[(wave * 16 + m16) * HSTRIDE];
        for (int k0 = 0; k0 < DD; k0 += 4) {
            const int kk = k0 + half * 2;
            float2 t2 = *(const float2*)(hrow + kk);
            v2f a; a.x = t2.x; a.y = t2.y;
            const float* bbase = &sW[(kk >> 1) * WPAIR + m16 * 2];
            #pragma unroll
            for (int t = 0; t < 8; ++t) {
                v2f b = *(const v2f*)(bbase + t * 32);
                acc[t] = __builtin_amdgcn_wmma_f32_16x16x4_f32(
                    false, a, false, b, (short)0, acc[t], false, false);
            }
        }
        const int rbase = row0 + wave * 16 + half * 8;
        #pragma unroll
        for (int t = 0; t < 8; ++t) {
            const int col  = t * 16 + m16;
            const float bb = b2[col];
            #pragma unroll
            for (int i = 0; i < 8; ++i) {
                const int row = rbase + i;
                if (row < nRows) Y[(size_t)row * DD + col] = acc[t][i] + bb;
            }
        }
    }
}

// ---------------------------------------------------------------------------
// COO SpMM scatter-add: y[r] += w * x[c]  (128 floats per edge)
// One wave32 per edge; lane l handles 4 consecutive floats (coalesced float4
// gather from L2-resident x; 4x global_atomic_add_f32 scatter).
// Accumulates ON TOP of existing y (the "+ local MLP" term is pre-stored).
// ---------------------------------------------------------------------------
__global__ __launch_bounds__(256) void spmm_atomic_kernel(
    const int* __restrict__ rows, const int* __restrict__ cols,
    const float* __restrict__ w, const float* __restrict__ x,
    float* __restrict__ y, int nEdges)
{
    const int tid  = blockIdx.x * blockDim.x + threadIdx.x;
    const int e    = tid >> 5;
    const int lane = tid & 31;
    if (e >= nEdges) return;
    const int   r  = rows[e];
    const int   c  = cols[e];
    const float ww = w[e];
    const float4 v = *(const float4*)(x + (size_t)c * DD + lane * 4);
    float* yp = y + (size_t)r * DD + lane * 4;
    atomicAdd(yp + 0, ww * v.x);
    atomicAdd(yp + 1, ww * v.y);
    atomicAdd(yp + 2, ww * v.z);
    atomicAdd(yp + 3, ww * v.w);
}

// ---------------------------------------------------------------------------
extern "C" void kernel_launch(void* const* d_in, const int* in_sizes, int n_in,
                              void* d_out, int out_size, void* d_ws, size_t ws_size,
                              hipStream_t stream)
{
    const float* feat = (const float*)d_in[0];
    const int*   er   = (const int*)d_in[1];
    const int*   ec   = (const int*)d_in[2];
    const float* ew   = (const float*)d_in[3];
    const float* f1W1 = (const float*)d_in[4];
    const float* f1b1 = (const float*)d_in[5];
    const float* f1W2 = (const float*)d_in[6];
    const float* f1b2 = (const float*)d_in[7];
    const float* f2W1 = (const float*)d_in[8];
    const float* f2b1 = (const float*)d_in[9];
    const float* f2W2 = (const float*)d_in[10];
    const float* f2b2 = (const float*)d_in[11];

    const size_t ND  = (size_t)NN * DD;
    const size_t DDd = (size_t)DD * DD;
    float* bufA = (float*)d_ws;       // message input h
    float* bufB = bufA + ND;          // x ping
    float* bufC = bufB + ND;          // x pong
    float* out  = (float*)d_out;

    const float* l0 = feat;
    const float* l1 = feat + ND;
    const float* l2 = feat + 2 * ND;

    const int mlpBlocks  = (NN + DD - 1) / DD;       // 782
    const int spmmBlocks = (EE * 32) / 256;          // 200000

    #define MLP(xin, Wset, j, yout)                                           \
        mlp_fused_kernel<<<mlpBlocks, 256, 0, stream>>>(                      \
            (xin), Wset##W1 + (size_t)(j) * DDd, Wset##b1 + (size_t)(j) * DD, \
            Wset##W2 + (size_t)(j) * DDd, Wset##b2 + (size_t)(j) * DD,        \
            (yout), NN)
    #define SPMM(i, xin, yout)                                                \
        spmm_atomic_kernel<<<spmmBlocks, 256, 0, stream>>>(                   \
            er + (size_t)(i) * EE, ec + (size_t)(i) * EE,                     \
            ew + (size_t)(i) * EE, (xin), (yout), EE)

    // i = 3: x = f1(2,l2) + A3 @ f1(3,l2)
    MLP(l2, f1, 3, bufA);
    MLP(l2, f1, 2, bufB);
    SPMM(3, bufA, bufB);
    // i = 2: x = f1(1,l1) + A2 @ f2(3,x)
    MLP(bufB, f2, 3, bufA);
    MLP(l1,  f1, 1, bufC);
    SPMM(2, bufA, bufC);
    // i = 1: x = f2(1,x) + A1 @ f2(2,x)   (both MLPs read old x in bufC)
    MLP(bufC, f2, 2, bufA);
    MLP(bufC, f2, 1, bufB);
    SPMM(1, bufA, bufB);
    // i = 0: x = f1(0,l0) + A0 @ f2(0,x)
    MLP(bufB, f2, 0, bufA);
    MLP(l0,  f1, 0, out);
    SPMM(0, bufA, out);

    #undef MLP
    #undef SPMM
}